// NerTr_86492051407176
// MI455X (gfx1250) — compile-verified
//
#include <hip/hip_runtime.h>

typedef __attribute__((ext_vector_type(2))) float v2f;
typedef __attribute__((ext_vector_type(4))) float v4f;
typedef __attribute__((ext_vector_type(8))) float v8f;

constexpr int BB = 32, SS = 512, DD = 768, CC = 25;
constexpr int TILE_S = 64;
constexpr int NTILES = SS / TILE_S;   // 8
constexpr int GROUPS = TILE_S / 16;   // 4
constexpr float LN_EPS = 1e-5f;

__device__ __forceinline__ float wave_sum(float v) {
  #pragma unroll
  for (int m = 16; m >= 1; m >>= 1) v += __shfl_xor(v, m, 32);
  return v;
}
__device__ __forceinline__ float wave_max(float v) {
  #pragma unroll
  for (int m = 16; m >= 1; m >>= 1) v = fmaxf(v, __shfl_xor(v, m, 32));
  return v;
}

// ---------------------------------------------------------------------------
// Fused main kernel: LN(bert) + bmm(cos_sim, LN(dec)) + add + LN + relu,
// then emb @ W_add via V_WMMA_F32_16X16X4_F32, then log_softmax -> ner_prob.
// One block = 64 sequence rows of one batch. 256 threads = 8 waves.
// ---------------------------------------------------------------------------
__global__ __launch_bounds__(256) void ner_main_kernel(
    const float* __restrict__ bert, const float* __restrict__ enc,
    const float* __restrict__ cos_sim, const float* __restrict__ dec_emb,
    const float* __restrict__ gamma, const float* __restrict__ beta,
    const float* __restrict__ Wadd, const float* __restrict__ badd,
    float* __restrict__ ner)
{
  __shared__ __align__(16) float sdec[CC][DD];       // 76800 B: LN(dec[b])
  __shared__ __align__(16) float semb[16][DD + 4];   // 49408 B: emb tile (pad 4 -> conflict-free WMMA A reads)
  __shared__ float slog[16][33];                     // logits tile
  __shared__ float scos[16][CC];                     // cos_sim rows of group
  __shared__ __align__(16) float sgam[DD];
  __shared__ __align__(16) float sbet[DD];

  const int b    = blockIdx.x / NTILES;
  const int tile = blockIdx.x % NTILES;
  const int s0   = tile * TILE_S;
  const int tid  = threadIdx.x;
  const int lane = tid & 31;
  const int wv   = tid >> 5;

  for (int i = tid; i < DD; i += 256) { sgam[i] = gamma[i]; sbet[i] = beta[i]; }
  __syncthreads();

  // ---- stage LN(decoder_embedding[b]) into LDS; wave wv does rows wv, wv+8, ...
  for (int c = wv; c < CC; c += 8) {
    const v4f* row = (const v4f*)(dec_emb + ((size_t)b * CC + c) * DD);
    float x[24]; float sum = 0.f;
    #pragma unroll
    for (int jj = 0; jj < 6; ++jj) {
      v4f t = row[jj * 32 + lane];
      x[jj*4+0]=t.x; x[jj*4+1]=t.y; x[jj*4+2]=t.z; x[jj*4+3]=t.w;
      sum += t.x + t.y + t.z + t.w;
    }
    float mean = wave_sum(sum) * (1.f / DD);
    float var = 0.f;
    #pragma unroll
    for (int q = 0; q < 24; ++q) { float d = x[q] - mean; var += d * d; }
    float rstd = rsqrtf(wave_sum(var) * (1.f / DD) + LN_EPS);
    v4f* orow = (v4f*)&sdec[c][0];
    #pragma unroll
    for (int jj = 0; jj < 6; ++jj) {
      int d0 = jj * 128 + lane * 4;
      v4f o;
      o.x = (x[jj*4+0]-mean)*rstd*sgam[d0+0] + sbet[d0+0];
      o.y = (x[jj*4+1]-mean)*rstd*sgam[d0+1] + sbet[d0+1];
      o.z = (x[jj*4+2]-mean)*rstd*sgam[d0+2] + sbet[d0+2];
      o.w = (x[jj*4+3]-mean)*rstd*sgam[d0+3] + sbet[d0+3];
      orow[jj * 32 + lane] = o;
    }
  }
  __syncthreads();

  for (int g = 0; g < GROUPS; ++g) {
    const int sg = s0 + g * 16;
    for (int t = tid; t < 16 * CC; t += 256)
      scos[t / CC][t % CC] = cos_sim[((size_t)(b * SS + sg + t / CC)) * CC + (t % CC)];
    __syncthreads();

    // ---- each wave computes 2 emb rows of the 16-row tile
    #pragma unroll
    for (int rr = 0; rr < 2; ++rr) {
      const int r = wv * 2 + rr;
      const int s = sg + r;
      const v4f* brow = (const v4f*)(bert + ((size_t)(b * SS + s)) * DD);
      const v4f* erow = (const v4f*)(enc  + ((size_t)(b * SS + s)) * DD);
      float x[24]; float sum = 0.f;
      #pragma unroll
      for (int jj = 0; jj < 6; ++jj) {
        v4f t = brow[jj * 32 + lane];
        x[jj*4+0]=t.x; x[jj*4+1]=t.y; x[jj*4+2]=t.z; x[jj*4+3]=t.w;
        sum += t.x + t.y + t.z + t.w;
      }
      float mean = wave_sum(sum) * (1.f / DD);
      float var = 0.f;
      #pragma unroll
      for (int q = 0; q < 24; ++q) { float d = x[q] - mean; var += d * d; }
      float rstd = rsqrtf(wave_sum(var) * (1.f / DD) + LN_EPS);
      float y[24];
      #pragma unroll
      for (int jj = 0; jj < 6; ++jj) {
        int d0 = jj * 128 + lane * 4;
        v4f e = erow[jj * 32 + lane];
        y[jj*4+0] = (x[jj*4+0]-mean)*rstd*sgam[d0+0] + sbet[d0+0] + e.x;
        y[jj*4+1] = (x[jj*4+1]-mean)*rstd*sgam[d0+1] + sbet[d0+1] + e.y;
        y[jj*4+2] = (x[jj*4+2]-mean)*rstd*sgam[d0+2] + sbet[d0+2] + e.z;
        y[jj*4+3] = (x[jj*4+3]-mean)*rstd*sgam[d0+3] + sbet[d0+3] + e.w;
      }
      // prob_query: y += cos_sim[b,s,:] @ LN(dec[b])
      for (int c = 0; c < CC; ++c) {
        float ccv = scos[r][c];
        const v4f* drow = (const v4f*)&sdec[c][0];
        #pragma unroll
        for (int jj = 0; jj < 6; ++jj) {
          v4f dv = drow[jj * 32 + lane];
          y[jj*4+0] += ccv * dv.x; y[jj*4+1] += ccv * dv.y;
          y[jj*4+2] += ccv * dv.z; y[jj*4+3] += ccv * dv.w;
        }
      }
      // second LN + relu -> semb
      float sum2 = 0.f;
      #pragma unroll
      for (int q = 0; q < 24; ++q) sum2 += y[q];
      float mean2 = wave_sum(sum2) * (1.f / DD);
      float var2 = 0.f;
      #pragma unroll
      for (int q = 0; q < 24; ++q) { float d = y[q] - mean2; var2 += d * d; }
      float rstd2 = rsqrtf(wave_sum(var2) * (1.f / DD) + LN_EPS);
      v4f* emrow = (v4f*)&semb[r][0];
      #pragma unroll
      for (int jj = 0; jj < 6; ++jj) {
        int d0 = jj * 128 + lane * 4;
        v4f o;
        o.x = fmaxf((y[jj*4+0]-mean2)*rstd2*sgam[d0+0] + sbet[d0+0], 0.f);
        o.y = fmaxf((y[jj*4+1]-mean2)*rstd2*sgam[d0+1] + sbet[d0+1], 0.f);
        o.z = fmaxf((y[jj*4+2]-mean2)*rstd2*sgam[d0+2] + sbet[d0+2], 0.f);
        o.w = fmaxf((y[jj*4+3]-mean2)*rstd2*sgam[d0+3] + sbet[d0+3], 0.f);
        emrow[jj * 32 + lane] = o;
      }
    }
    __syncthreads();

    // ---- wave 0: logits[16x25] = emb[16x768] @ W_add[768x25] via WMMA f32 16x16x4
    if (wv == 0) {
      const int mrow = lane & 15;
      const int kh   = (lane >> 4) << 1;   // 0 or 2: ISA A-layout K = 2*(lane/16)+v
      const int n1   = 16 + mrow;
      // branch-free tail handling: clamp column index, zero via loop-invariant mask
      const float msk = (n1 < CC) ? 1.f : 0.f;
      const int   n1c = (n1 < CC) ? n1 : 0;
      v8f acc0 = {0.f,0.f,0.f,0.f,0.f,0.f,0.f,0.f};
      v8f acc1 = {0.f,0.f,0.f,0.f,0.f,0.f,0.f,0.f};
      for (int k = 0; k < DD; k += 4) {
        v2f a;  a.x  = semb[mrow][k + kh];               a.y  = semb[mrow][k + kh + 1];
        v2f b0; b0.x = Wadd[(k + kh) * CC + mrow];       b0.y = Wadd[(k + kh + 1) * CC + mrow];
        v2f b1; b1.x = msk * Wadd[(k + kh) * CC + n1c];  b1.y = msk * Wadd[(k + kh + 1) * CC + n1c];
        acc0 = __builtin_amdgcn_wmma_f32_16x16x4_f32(false, a, false, b0, (short)0, acc0, false, false);
        acc1 = __builtin_amdgcn_wmma_f32_16x16x4_f32(false, a, false, b1, (short)0, acc1, false, false);
      }
      const int rbase = (lane >> 4) * 8;   // D layout: VGPR r -> M = r + 8*(lane/16)
      #pragma unroll
      for (int r8 = 0; r8 < 8; ++r8) {
        slog[rbase + r8][mrow]      = acc0[r8];
        slog[rbase + r8][16 + mrow] = acc1[r8];
      }
    }
    __syncthreads();

    // ---- log_softmax per row, write ner_prob
    if (tid < 16) {
      const int s = sg + tid;
      float z[CC]; float m = -1e30f;
      #pragma unroll
      for (int n = 0; n < CC; ++n) { z[n] = slog[tid][n] + badd[n]; m = fmaxf(m, z[n]); }
      float se = 0.f;
      #pragma unroll
      for (int n = 0; n < CC; ++n) se += __expf(z[n] - m);
      float lse = m + __logf(se);
      float* orow = ner + ((size_t)(b * SS + s)) * CC;
      #pragma unroll
      for (int n = 0; n < CC; ++n) orow[n] = z[n] - lse;
    }
    __syncthreads();
  }
}

// ---------------------------------------------------------------------------
// CRF negative log-likelihood (forward algorithm). One wave32 per batch.
// Emissions staged in LDS; logsumexp factored via precomputed exp(trans).
// ---------------------------------------------------------------------------
__global__ __launch_bounds__(32) void crf_nll_kernel(
    const float* __restrict__ ner, const int* __restrict__ labels,
    const float* __restrict__ startt, const float* __restrict__ endt,
    const float* __restrict__ trans, float* __restrict__ llh)
{
  __shared__ float sem[SS][CC];   // 51200 B
  __shared__ float setr[CC][CC];  // exp(trans)
  __shared__ float sexp[32];
  const int b = blockIdx.x, t = threadIdx.x;
  for (int i = t; i < SS * CC; i += 32) (&sem[0][0])[i] = ner[(size_t)b * SS * CC + i];
  for (int i = t; i < CC * CC; i += 32) (&setr[0][0])[i] = __expf(trans[i]);
  __syncthreads();

  // gold-path score (emissions + transitions), parallel over time
  float num = 0.f;
  for (int s = t; s < SS; s += 32) {
    int tg = labels[b * SS + s];
    num += sem[s][tg];
    if (s + 1 < SS) num += trans[tg * CC + labels[b * SS + s + 1]];
  }
  num = wave_sum(num);

  // forward recursion
  float scr = (t < CC) ? (startt[t] + sem[0][t]) : -1e30f;
  for (int s = 1; s < SS; ++s) {
    float m = wave_max(scr);
    sexp[t] = __expf(scr - m);
    __syncthreads();
    if (t < CC) {
      float acc = 0.f;
      #pragma unroll
      for (int i = 0; i < CC; ++i) acc += sexp[i] * setr[i][t];
      scr = m + __logf(acc) + sem[s][t];
    }
    __syncthreads();
  }
  float fin = (t < CC) ? scr + endt[t] : -1e30f;
  float m2 = wave_max(fin);
  float se = wave_sum((t < CC) ? __expf(fin - m2) : 0.f);
  if (t == 0) {
    float logZ = m2 + __logf(se);
    num += startt[labels[b * SS]] + endt[labels[b * SS + SS - 1]];
    llh[b] = num - logZ;
  }
}

__global__ void loss_kernel(const float* __restrict__ llh, float* __restrict__ out) {
  if (threadIdx.x == 0 && blockIdx.x == 0) {
    float s = 0.f;
    for (int b = 0; b < BB; ++b) s += llh[b];
    out[0] = -(s / (float)BB);
  }
}

// ---------------------------------------------------------------------------
// Viterbi decode on softmax(cos_sim). One wave32 per batch; emissions and
// byte back-pointers fully in LDS so backtrack never leaves the WGP.
// ---------------------------------------------------------------------------
__global__ __launch_bounds__(32) void viterbi_kernel(
    const float* __restrict__ cos_sim, const float* __restrict__ startt,
    const float* __restrict__ endt, const float* __restrict__ trans,
    float* __restrict__ path)
{
  __shared__ float sp[SS][CC];                 // 51200 B
  __shared__ float str[CC][CC];
  __shared__ unsigned char hist[SS - 1][CC];   // 12775 B
  __shared__ float sc[2][32];
  const int b = blockIdx.x, t = threadIdx.x;
  for (int i = t; i < SS * CC; i += 32) (&sp[0][0])[i] = cos_sim[(size_t)b * SS * CC + i];
  for (int i = t; i < CC * CC; i += 32) (&str[0][0])[i] = trans[i];
  __syncthreads();

  // softmax rows in place
  for (int s = t; s < SS; s += 32) {
    float m = -1e30f;
    for (int n = 0; n < CC; ++n) m = fmaxf(m, sp[s][n]);
    float se = 0.f;
    for (int n = 0; n < CC; ++n) se += __expf(sp[s][n] - m);
    float inv = 1.f / se;
    for (int n = 0; n < CC; ++n) sp[s][n] = __expf(sp[s][n] - m) * inv;
  }
  __syncthreads();

  int cur = 0;
  sc[0][t] = (t < CC) ? startt[t] + sp[0][t] : -1e30f;
  __syncthreads();
  for (int s = 1; s < SS; ++s) {
    if (t < CC) {
      float best = -1e30f; int bi = 0;
      #pragma unroll
      for (int i = 0; i < CC; ++i) {
        float v = sc[cur][i] + str[i][t];
        if (v > best) { best = v; bi = i; }
      }
      sc[cur ^ 1][t] = best + sp[s][t];
      hist[s - 1][t] = (unsigned char)bi;
    } else {
      sc[cur ^ 1][t] = -1e30f;
    }
    __syncthreads();
    cur ^= 1;
  }
  if (t == 0) {
    float best = -1e30f; int bi = 0;
    for (int j = 0; j < CC; ++j) {
      float v = sc[cur][j] + endt[j];
      if (v > best) { best = v; bi = j; }
    }
    int tag = bi;
    path[b * SS + SS - 1] = (float)tag;
    for (int s = SS - 2; s >= 0; --s) { tag = hist[s][tag]; path[b * SS + s] = (float)tag; }
  }
}

extern "C" void kernel_launch(void* const* d_in, const int* in_sizes, int n_in,
                              void* d_out, int out_size, void* d_ws, size_t ws_size,
                              hipStream_t stream) {
  (void)in_sizes; (void)n_in; (void)out_size; (void)ws_size;
  const float* bert   = (const float*)d_in[0];
  const float* enc    = (const float*)d_in[1];
  const float* cosim  = (const float*)d_in[2];
  const float* dec    = (const float*)d_in[3];
  const int*   labels = (const int*)  d_in[4];
  const float* gamma  = (const float*)d_in[5];
  const float* beta   = (const float*)d_in[6];
  const float* Wadd   = (const float*)d_in[7];
  const float* badd   = (const float*)d_in[8];
  const float* startt = (const float*)d_in[9];
  const float* endt   = (const float*)d_in[10];
  const float* trans  = (const float*)d_in[11];
  float* out = (float*)d_out;
  float* ner = (float*)d_ws;                       // [B,S,C] log-probs
  float* llh = ner + (size_t)BB * SS * CC;         // [B]

  ner_main_kernel<<<BB * NTILES, 256, 0, stream>>>(bert, enc, cosim, dec,
                                                   gamma, beta, Wadd, badd, ner);
  crf_nll_kernel<<<BB, 32, 0, stream>>>(ner, labels, startt, endt, trans, llh);
  loss_kernel<<<1, 32, 0, stream>>>(llh, out);
  viterbi_kernel<<<BB, 32, 0, stream>>>(cosim, startt, endt, trans, out + 1);
}